// RadiusKnnGraph_89816356094632
// MI455X (gfx1250) — compile-verified
//
#include <hip/hip_runtime.h>
#include <hip/hip_bf16.h>

// Radius-KNN graph for MI455X (gfx1250, wave32).
//
// d2(i,j) = sq_i + sq_j - 2*x_i.x_j computed as ONE v_wmma_f32_16x16x4_f32:
//   A row i  = (x_i0, x_i1, x_i2, sq_i)        (16x4)
//   B col j  = (-2*x_j0, -2*x_j1, -2*x_j2, 1)  (4x16)
//   C[i][j]  = sq_j
//   D = A*B + C = sq_i + sq_j - 2*x_i.x_j
//
// batch is sorted -> same-batch candidates are one contiguous index range:
//  * a prefix kernel builds batch_start[17]
//  * each wave scans only the tile span covering its rows' batches
//  * per-lane the batch test is a pure-VALU range check (no loads in the
//    candidate loop at all)

typedef __attribute__((ext_vector_type(2))) float v2f;
typedef __attribute__((ext_vector_type(8))) float v8f;

#define N_PTS   8192
#define KNN     32
#define RAD2    0.0625f   // 0.25^2, exact in fp32
#define NBATCH  16

// ---------------------------------------------------------------------------
// batch_start[b] = first index i with batch[i] >= b, b = 0..16.
// ---------------------------------------------------------------------------
__global__ void RadiusKnn_batch_start_kernel(const int* __restrict__ batch,
                                             int* __restrict__ bstart) {
    int b = threadIdx.x;
    if (b > NBATCH) return;
    int lo = 0, hi = N_PTS;
    while (lo < hi) {
        int mid = (lo + hi) >> 1;
        if (batch[mid] < b) lo = mid + 1; else hi = mid;
    }
    bstart[b] = lo;
}

// ---------------------------------------------------------------------------
// Sorted top-32 insert (ascending by (d2, idx)); caller guarantees
// cd < ld[KNN-1]. Strict '<' implements lax.top_k tie-breaking: equal
// distances keep the earlier (lower-index) entry.
// ---------------------------------------------------------------------------
__device__ __forceinline__ void insert32(float (&ld)[KNN], int (&li)[KNN],
                                         float cd, int ci) {
#pragma unroll
    for (int k = KNN - 1; k > 0; --k) {
        bool  sh   = cd < ld[k - 1];             // shift down
        bool  here = !sh && (cd < ld[k]);        // insert here
        float nd = sh ? ld[k - 1] : (here ? cd : ld[k]);
        int   ni = sh ? li[k - 1] : (here ? ci : li[k]);
        ld[k] = nd; li[k] = ni;
    }
    if (cd < ld[0]) { ld[0] = cd; li[0] = ci; }
}

// ---------------------------------------------------------------------------
// Main kernel: one wave (32 lanes) per 32 target rows.
// ---------------------------------------------------------------------------
__global__ __launch_bounds__(32)
void RadiusKnn_graph_kernel(const float* __restrict__ x,
                            const int*   __restrict__ batch,
                            const int*   __restrict__ bstart,
                            int*         __restrict__ out) {
    __shared__ float sx[32][3];   // this wave's 32 target points
    __shared__ float ssq[32];     // their squared norms
    __shared__ float sd[32][17];  // 32x16 d2 tile, padded stride 17 (banks)

    const int  lane    = threadIdx.x;            // 0..31
    const int  base    = blockIdx.x * 32;
    const int  row     = base + lane;            // this lane's target row
    const bool hi_half = lane >= 16;
    const int  m       = lane & 15;

    // --- load own target point, share via LDS -----------------------------
    float x0 = x[row * 3 + 0];
    float x1 = x[row * 3 + 1];
    float x2 = x[row * 3 + 2];
    float sq = x0 * x0 + x1 * x1 + x2 * x2;      // same assoc as jnp.sum(x*x)
    sx[lane][0] = x0; sx[lane][1] = x1; sx[lane][2] = x2;
    ssq[lane] = sq;
    const int mybatch = batch[row];
    // per-lane same-batch index range (replaces batch[] loads in hot loop)
    const int my_jlo = bstart[mybatch];
    const int my_jhi = bstart[mybatch + 1];
    __syncthreads();   // single-wave workgroup: barrier lowers to S_NOP

    // --- wave-uniform source-tile range from sorted batch -----------------
    const int bmin = batch[base];
    const int bmax = batch[base + 31];
    const int jlo  = bstart[bmin];
    const int jhi  = bstart[bmax + 1];
    const int t0   = jlo >> 4;
    const int t1   = (jhi + 15) >> 4;            // exclusive tile bound

    // --- A operands (16x4 f32 layout: low lanes K=0,1; high lanes K=2,3) --
    // Tile0 = target rows base+0..15, Tile1 = base+16..31.
    v2f a0, a1;
    if (!hi_half) {
        a0[0] = sx[m][0];      a0[1] = sx[m][1];
        a1[0] = sx[16 + m][0]; a1[1] = sx[16 + m][1];
    } else {
        a0[0] = sx[m][2];      a0[1] = ssq[m];
        a1[0] = sx[16 + m][2]; a1[1] = ssq[16 + m];
    }

    // --- per-lane top-32 list (registers, sorted ascending by (d2, idx)) --
    float ld[KNN];
    int   li[KNN];
#pragma unroll
    for (int k = 0; k < KNN; ++k) { ld[k] = __builtin_inff(); li[k] = -1; }

    // --- scan source tiles -------------------------------------------------
    for (int ct = t0; ct < t1; ++ct) {           // wave-uniform loop
        const int colb = ct << 4;
        const int j    = colb + m;               // this lane's column in tile

        float y0 = x[j * 3 + 0];                 // global_load_b96, L2-hot
        float y1 = x[j * 3 + 1];
        float y2 = x[j * 3 + 2];
        float sqj = y0 * y0 + y1 * y1 + y2 * y2;

        // B operand (4x16): low lanes rows K=0,1; high lanes rows K=2,3
        v2f b;
        if (!hi_half) { b[0] = -2.0f * y0; b[1] = -2.0f * y1; }
        else          { b[0] = -2.0f * y2; b[1] = 1.0f; }

        v8f c;
#pragma unroll
        for (int r = 0; r < 8; ++r) c[r] = sqj;  // C[i][j] = sq_j

        // D = A*B + C  -> exact 16x16 block of squared distances
        v8f d0 = __builtin_amdgcn_wmma_f32_16x16x4_f32(
                     false, a0, false, b, (short)0, c, false, false);
        v8f d1 = __builtin_amdgcn_wmma_f32_16x16x4_f32(
                     false, a1, false, b, (short)0, c, false, false);

        // scatter WMMA (col-lane) layout -> row-major LDS tile
        // D[r][lane]: target row = r + 8*hi_half (within 16-row tile), col m
#pragma unroll
        for (int r = 0; r < 8; ++r) {
            int mrow = r + (hi_half ? 8 : 0);
            sd[mrow][m]      = d0[r];
            sd[16 + mrow][m] = d1[r];
        }
        __syncthreads();

        // selection: lane owns target row `row`; scan 16 candidates in
        // ascending index order (required for exact top_k tie-break).
        // Paired reads -> ds_load_2addr_b32 fusion, one dscnt wait per pair.
        for (int n2 = 0; n2 < 16; n2 += 2) {
            const float da = sd[lane][n2];
            const float db = sd[lane][n2 + 1];
            const int   ja = colb + n2;
            const int   jb = ja + 1;

            const bool oka = (ja >= my_jlo) & (ja < my_jhi) & (ja != row)
                             & (da <= RAD2);
            if (oka && da < ld[KNN - 1]) insert32(ld, li, da, ja);

            const bool okb = (jb >= my_jlo) & (jb < my_jhi) & (jb != row)
                             & (db <= RAD2);
            if (okb && db < ld[KNN - 1]) insert32(ld, li, db, jb);
        }
        __syncthreads();   // protect sd before next tile overwrites it
    }

    // --- emit edges: row 0 = sources, row 1 = targets, -1 padding ---------
    const int obase = row * KNN;
#pragma unroll
    for (int k = 0; k < KNN; ++k) {
        int s = li[k];
        out[obase + k]               = s;
        out[N_PTS * KNN + obase + k] = (s >= 0) ? row : -1;
    }
}

// ---------------------------------------------------------------------------
extern "C" void kernel_launch(void* const* d_in, const int* in_sizes, int n_in,
                              void* d_out, int out_size, void* d_ws, size_t ws_size,
                              hipStream_t stream) {
    (void)in_sizes; (void)n_in; (void)out_size; (void)ws_size;
    const float* x      = (const float*)d_in[0];
    const int*   batch  = (const int*)d_in[1];
    int*         bstart = (int*)d_ws;            // 17 ints of scratch
    int*         out    = (int*)d_out;           // int32 edge_index [2, N*K]

    RadiusKnn_batch_start_kernel<<<1, 32, 0, stream>>>(batch, bstart);
    RadiusKnn_graph_kernel<<<N_PTS / 32, 32, 0, stream>>>(x, batch, bstart, out);
}